// SimpleClassificationRNN_46772193853594
// MI455X (gfx1250) — compile-verified
//
#include <hip/hip_runtime.h>
#include <math.h>

// RNN recurrence with TRANSPOSED state G = H^T (hidden rows, batch cols):
//   G_{t+1} = tanh( W_hh^T @ G_t + w_ih^T x_t^T + bias )
// One 16-batch tile per wave32. Matmul = two parallel chains of
// V_WMMA_F32_16X16X4_F32 (W_hh^T as constant A operand). The D-layout ->
// B-operand-layout state conversion is only lane-half swaps: ideally 4x
// v_permlane16_swap_b32, else v_permlanex16 + cndmask. Input projection and
// bias ride the C operand via per-lane FMAs (off the critical path). No LDS.

#define RNN_HIDDEN 16

typedef __attribute__((ext_vector_type(2))) float v2f;
typedef __attribute__((ext_vector_type(8))) float v8f;
typedef __attribute__((ext_vector_type(2))) unsigned int v2u;

#if defined(__has_builtin)
# if __has_builtin(__builtin_amdgcn_tanhf)
#  define FAST_TANH(x) __builtin_amdgcn_tanhf(x)
# else
#  define FAST_TANH(x) tanhf(x)
# endif
#else
# define FAST_TANH(x) tanhf(x)
#endif

// Swap lane halves (lane i <-> lane i^16), identity mapping inside each half.
// old operand = 0: never read when all lanes are active (fi=bc=false).
__device__ __forceinline__ float halfswap(float v) {
#if defined(__has_builtin) && __has_builtin(__builtin_amdgcn_permlanex16)
    int i = __builtin_amdgcn_permlanex16(0, __float_as_int(v),
                                         0x76543210, (int)0xfedcba98u, false, false);
    return __int_as_float(i);
#else
    return __shfl_xor(v, 16, 32);
#endif
}

// Produce lolo = {a.lo | b.lo in hi lanes}, hihi = {a.hi in lo lanes | b.hi}.
__device__ __forceinline__ void halfswap_pair(float a, float b, bool ishi,
                                              float& lolo, float& hihi) {
#if defined(__has_builtin) && __has_builtin(__builtin_amdgcn_permlane16_swap)
    // v_permlane16_swap_b32 vdst, src swaps vdst.lanes[0:15] <-> src.lanes[16:31].
    // vdst=b, src=a: new_vdst = {a.hi, b.hi}, new_src = {a.lo, b.lo}.
    (void)ishi;
    v2u r = __builtin_amdgcn_permlane16_swap(__float_as_uint(b), __float_as_uint(a),
                                             false, false);
    hihi = __uint_as_float(r[0]);
    lolo = __uint_as_float(r[1]);
#else
    const float sa = halfswap(a);   // a.hi -> lo lanes, a.lo -> hi lanes
    const float sb = halfswap(b);
    lolo = ishi ? sb : a;           // lo: a.lo ; hi: b.lo
    hihi = ishi ? b : sa;           // lo: a.hi ; hi: b.hi
#endif
}

__global__ __launch_bounds__(32)
void rnn_wmma_f32_kernel(const float* __restrict__ seq,    // [B, T]
                         const float* __restrict__ w_ih,   // [1, 16]
                         const float* __restrict__ w_hh,   // [16, 16]
                         const float* __restrict__ b_ih,   // [16]
                         const float* __restrict__ b_hh,   // [16]
                         const float* __restrict__ lin_w,  // [1, 16]
                         const float* __restrict__ lin_b,  // [1]
                         float* __restrict__ out,          // [B] logits, then [B,16] hc
                         int B, int T)
{
    const int lane = threadIdx.x;     // 0..31, wave32
    const int nl   = lane & 15;       // batch column within tile
    const int hi   = lane >> 4;       // lane half
    const bool ishi = (hi != 0);
    const int m0   = blockIdx.x * 16; // tile's first batch row

    // ---- constants held in registers for the whole recurrence ----
    // A operand = W_hh^T in A-layout: chunk c, VGPR0 = K=4c(+2 hi), VGPR1 = next K.
    // A(m, k) = W_hh^T[m][k] = w_hh[k*16 + m], m = nl on lanes.
    v2f Aw[4];
    #pragma unroll
    for (int c = 0; c < 4; ++c) {
        const int k = 4 * c + 2 * hi;
        Aw[c].x = w_hh[k * 16 + nl];
        Aw[c].y = w_hh[(k + 1) * 16 + nl];
    }
    // Bias and input-projection row vectors per lane component: row = j + 8*hi.
    v8f cbias, wihr;
    #pragma unroll
    for (int j = 0; j < 8; ++j) {
        const int r = j + 8 * hi;
        cbias[j] = b_ih[r] + b_hh[r];
        wihr[j]  = w_ih[r];
    }
    v8f zero8;
    #pragma unroll
    for (int j = 0; j < 8; ++j) zero8[j] = 0.0f;

    // state G in WMMA C/D layout: h[j] = G[row j (lo) / j+8 (hi)][col nl]
    v8f h = zero8;

    // both lane halves load x[b = nl] (same address)
    const float* seq_row = seq + (size_t)(m0 + nl) * (size_t)T;

    #pragma unroll 4
    for (int t = 0; t < T; ++t) {
        const float xv = seq_row[t];

        // C operand = bias + w_ih (X) x  (rank-1 input term, off critical path)
        v8f cin;
        #pragma unroll
        for (int j = 0; j < 8; ++j) cin[j] = __builtin_fmaf(wihr[j], xv, cbias[j]);

        // D-layout -> B-operand-layout: lane-half swaps only.
        float b0x, b0y, b1x, b1y, b2x, b2y, b3x, b3y;
        halfswap_pair(h[0], h[2], ishi, b0x, b2x);  // rows 0|2 , 8|10
        halfswap_pair(h[1], h[3], ishi, b0y, b2y);  // rows 1|3 , 9|11
        halfswap_pair(h[4], h[6], ishi, b1x, b3x);  // rows 4|6 , 12|14
        halfswap_pair(h[5], h[7], ishi, b1y, b3y);  // rows 5|7 , 13|15
        v2f B0, B1, B2, B3;
        B0.x = b0x; B0.y = b0y;
        B1.x = b1x; B1.y = b1y;
        B2.x = b2x; B2.y = b2y;
        B3.x = b3x; B3.y = b3y;

        // Two parallel fp32 WMMA chains (depth 2 instead of 4)
        v8f acc1 = __builtin_amdgcn_wmma_f32_16x16x4_f32(false, Aw[0], false, B0, (short)0, cin,   false, false);
        v8f acc2 = __builtin_amdgcn_wmma_f32_16x16x4_f32(false, Aw[2], false, B2, (short)0, zero8, false, false);
        acc1 = __builtin_amdgcn_wmma_f32_16x16x4_f32(false, Aw[1], false, B1, (short)0, acc1, false, false);
        acc2 = __builtin_amdgcn_wmma_f32_16x16x4_f32(false, Aw[3], false, B3, (short)0, acc2, false, false);

        #pragma unroll
        for (int j = 0; j < 8; ++j) h[j] = FAST_TANH(acc1[j] + acc2[j]);
    }

    // ---- epilogue, no LDS ----
    // hc[b][k] = G[k][b]: each lane's 8 components are CONTIGUOUS in d_out.
    {
        float* hc = out + (size_t)B + (size_t)(m0 + nl) * RNN_HIDDEN + 8 * hi;
        #pragma unroll
        for (int j = 0; j < 8; ++j) hc[j] = h[j];
    }
    // logits: z[b] = sum_k G[k][b] * lin_w[k] + lin_b; half-swap reduction.
    {
        float part = 0.0f;
        #pragma unroll
        for (int j = 0; j < 8; ++j) part += h[j] * lin_w[j + 8 * hi];
        const float other = halfswap(part);
        if (!ishi) {
            const float z = part + other + lin_b[0];
            out[m0 + nl] = 1.0f / (1.0f + __expf(-z));
        }
    }
}

extern "C" void kernel_launch(void* const* d_in, const int* in_sizes, int n_in,
                              void* d_out, int out_size, void* d_ws, size_t ws_size,
                              hipStream_t stream) {
    (void)n_in; (void)d_ws; (void)ws_size;
    const float* seq   = (const float*)d_in[0];
    const float* w_ih  = (const float*)d_in[1];
    const float* w_hh  = (const float*)d_in[2];
    const float* b_ih  = (const float*)d_in[3];
    const float* b_hh  = (const float*)d_in[4];
    const float* lin_w = (const float*)d_in[5];
    const float* lin_b = (const float*)d_in[6];
    float* out = (float*)d_out;

    // out_size = B (logits) + B*HIDDEN (final state) = 17*B
    const int B = out_size / (RNN_HIDDEN + 1);
    const int T = in_sizes[0] / B;

    dim3 grid(B / 16), block(32);
    rnn_wmma_f32_kernel<<<grid, block, 0, stream>>>(seq, w_ih, w_hh, b_ih, b_hh,
                                                    lin_w, lin_b, out, B, T);
}